// SpikingLayer_72292889526608
// MI455X (gfx1250) — compile-verified
//
#include <hip/hip_runtime.h>

typedef __attribute__((ext_vector_type(2))) float v2f;
typedef __attribute__((ext_vector_type(8))) float v8f;
typedef __attribute__((ext_vector_type(4))) unsigned int u32x4;
typedef __attribute__((ext_vector_type(8))) int i32x8;
typedef __attribute__((ext_vector_type(4))) int i32x4;

namespace {
constexpr int kB = 256, kT = 250, kC = 700, kD = 512;
constexpr int WAVES = 4;            // 4 waves -> one wave per SIMD32 of the WGP
constexpr int DTILE = WAVES * 16;   // 64 neuron columns per workgroup
constexpr int KS = kC;              // LDS row stride (700 % 64 == 60 -> conflict-free b64 frag loads)
constexpr float kTref = 5.0f;
constexpr int BTD = kB * kT * kD;
}

// Issue a TDM load of x[b0:b0+16, t, 0:700] (fp32, row stride T*C) into LDS.
// D# layout per CDNA5 ISA ch.8: group0 = {flags, lds_addr, global_addr}, group1 = dims.
__device__ __forceinline__ void tdm_load_x_tile(const float* gsrc, unsigned lds_byte_off)
{
    unsigned long long ga = (unsigned long long)(uintptr_t)gsrc;
    u32x4 g0;
    g0.x = 0x1u;                                   // count=1 (valid user descriptor)
    g0.y = lds_byte_off;                           // lds_addr (bytes)
    g0.z = (unsigned)(ga & 0xffffffffu);           // global_addr[31:0]
    g0.w = (unsigned)((ga >> 32) & 0x1ffffffu)     // global_addr[56:32]
         | (2u << 30);                             // type=2 ("image")
    i32x8 g1;
    g1[0] = (int)(2u << 16);                       // data_size = 2 -> 4 bytes/elem
    g1[1] = (int)((unsigned)kC << 16);             // tensor_dim0[15:0] = 700  (bits 63:48)
    g1[2] = (int)(16u << 16);                      // tensor_dim0[31:16]=0 | tensor_dim1[15:0]=16
    g1[3] = (int)((unsigned)kC << 16);             // tensor_dim1[31:16]=0 | tile_dim0 = 700
    g1[4] = 16;                                    // tile_dim1 = 16, tile_dim2 = 0
    g1[5] = kT * kC;                               // tensor_dim0_stride[31:0] = 175000 elems
    g1[6] = 0;                                     // stride[47:32]=0, tensor_dim1_stride lo = 0
    g1[7] = 0;
    i32x4 gz4 = {0, 0, 0, 0};                      // groups 2/3 unused (2D tensor)
    i32x8 gz8 = {0, 0, 0, 0, 0, 0, 0, 0};          // extra group (clang-23 form), unused
    __builtin_amdgcn_tensor_load_to_lds(g0, g1, gz4, gz4, gz8, 0);
}

__global__ __launch_bounds__(WAVES * 32, 1)
void snn_fused(const float* __restrict__ xg, const float* __restrict__ wg,
               const float* __restrict__ alpha, const float* __restrict__ beta,
               const float* __restrict__ thv, float* __restrict__ out)
{
    __shared__ float wlds[DTILE * KS];   // w^T stripe: [dlocal][c], loaded once (179.2 KB)
    __shared__ float xs[2][16 * KS];     // double-buffered x tile: [brow][c] (2 x 44.8 KB)

    const int tid  = threadIdx.x;
    const int wave = tid >> 5;
    const int lane = tid & 31;
    const int hf   = lane >> 4;   // half-wave select (K/M split per ISA layout)
    const int row  = lane & 15;

    const int b0 = blockIdx.x * 16;
    const int d0 = blockIdx.y * DTILE;
    const int dl = wave * 16 + row;   // local neuron column for B fragments
    const int d  = d0 + dl;           // this lane's output column

    // ---- one-time: stage w[:, d0:d0+64] transposed into LDS as [dlocal][c] ----
    {
        const int dc = tid & (DTILE - 1);          // 0..63 -> 256B coalesced global reads
        for (int c = (tid >> 6); c < kC; c += 2) {
            wlds[dc * KS + c] = wg[c * kD + d0 + dc];
        }
    }

    // ---- prologue: TDM the t=0 x tile into buffer 0 (wave 0 issues; EXEC ignored) ----
    if (wave == 0) {
        tdm_load_x_tile(xg + (size_t)b0 * kT * kC,
                        (unsigned)(uintptr_t)&xs[0][0]);
    }

    const float alpha_d = alpha[d];
    const float beta_d  = beta[d];
    const float th_d    = thv[d];
    const float ombeta  = 1.0f - beta_d;

    // LIF state: 8 (b,d) cells per lane, matching the WMMA f32 C/D layout
    float syn[8], mem[8], ref[8];
    #pragma unroll
    for (int v = 0; v < 8; ++v) { syn[v] = 0.f; mem[v] = 0.f; ref[v] = 0.f; }

    // t = 0 outputs are the zero initial state
    #pragma unroll
    for (int v = 0; v < 8; ++v) {
        const int b   = b0 + hf * 8 + v;
        const int off = (b * kT + 0) * kD + d;
        __builtin_nontemporal_store(0.f, out + off);
        __builtin_nontemporal_store(0.f, out + BTD + off);
        __builtin_nontemporal_store(0.f, out + 2 * BTD + off);
    }

    for (int t = 0; t < kT - 1; ++t) {
        __builtin_amdgcn_s_wait_tensorcnt(0);  // wave0: tile t landed; others: no-op
        __syncthreads();                       // tile t visible WG-wide; buf[t^1] free

        // overlap: DMA the next x tile into the other buffer while we compute
        if (t < kT - 2 && wave == 0) {
            tdm_load_x_tile(xg + ((size_t)b0 * kT + (t + 1)) * kC,
                            (unsigned)(uintptr_t)&xs[(t + 1) & 1][0]);
        }

        // ---- h tile: 175 x v_wmma_f32_16x16x4_f32, two accumulators for ILP ----
        v8f acc0 = {0.f,0.f,0.f,0.f,0.f,0.f,0.f,0.f};
        v8f acc1 = {0.f,0.f,0.f,0.f,0.f,0.f,0.f,0.f};
        // A frag (16x4 f32): lanes 0-15 hold K=k..k+1 of their M row, lanes 16-31 K=k+2..k+3
        const float* ap = &xs[t & 1][0] + row * KS + 2 * hf;
        const float* bp = wlds + dl * KS + 2 * hf;
        #pragma unroll 3
        for (int k = 0; k < 696; k += 8) {
            v2f a0 = *(const v2f*)(ap + k);
            v2f w0 = *(const v2f*)(bp + k);
            acc0 = __builtin_amdgcn_wmma_f32_16x16x4_f32(false, a0, false, w0,
                                                         (short)0, acc0, false, false);
            v2f a1 = *(const v2f*)(ap + k + 4);
            v2f w1 = *(const v2f*)(bp + k + 4);
            acc1 = __builtin_amdgcn_wmma_f32_16x16x4_f32(false, a1, false, w1,
                                                         (short)0, acc1, false, false);
        }
        {   // K tail: 700 = 87*8 + 4
            v2f a0 = *(const v2f*)(ap + 696);
            v2f w0 = *(const v2f*)(bp + 696);
            acc0 = __builtin_amdgcn_wmma_f32_16x16x4_f32(false, a0, false, w0,
                                                         (short)0, acc0, false, false);
        }

        // ---- LIF update + streaming output (t+1) ----
        #pragma unroll
        for (int v = 0; v < 8; ++v) {
            const float h    = acc0[v] + acc1[v];
            const bool  c    = (mem[v] - th_d) > 0.0f;
            const float spk  = c ? 1.0f : 0.0f;
            float       r    = c ? kTref : ref[v];
            const float nsyn = alpha_d * syn[v] + h;
            float       nmem = beta_d * mem[v] + ombeta * syn[v] - spk * th_d;
            nmem = (r < 1.0f) ? nmem : 0.0f;   // refractory gate (after ref<-TREF)
            r    = (r > 0.0f) ? r - 1.0f : r;

            const int b   = b0 + hf * 8 + v;
            const int off = (b * kT + (t + 1)) * kD + d;
            __builtin_nontemporal_store(nsyn, out + off);
            __builtin_nontemporal_store(nmem, out + BTD + off);
            __builtin_nontemporal_store(spk,  out + 2 * BTD + off);
            syn[v] = nsyn; mem[v] = nmem; ref[v] = r;
        }
    }
}

extern "C" void kernel_launch(void* const* d_in, const int* in_sizes, int n_in,
                              void* d_out, int out_size, void* d_ws, size_t ws_size,
                              hipStream_t stream) {
    const float* x     = (const float*)d_in[0];
    const float* w     = (const float*)d_in[1];
    const float* alpha = (const float*)d_in[2];
    const float* beta  = (const float*)d_in[3];
    const float* th    = (const float*)d_in[4];
    float* out = (float*)d_out;

    dim3 grid(kB / 16, kD / DTILE);   // (16, 8) = 128 workgroups, 4 waves each
    snn_fused<<<grid, WAVES * 32, 0, stream>>>(x, w, alpha, beta, th, out);
}